// EdgeGATLayer_2422361555737
// MI455X (gfx1250) — compile-verified
//
#include <hip/hip_runtime.h>
#include <hip/hip_bf16.h>

#define DD 64
#define K3D 192
#define SLOPE 0.01f

// chunked LDS layouts (dword units)
// A (16 rows x 192 K): idx = (k>>4)*A_CH + ((k>>1)&1)*A_H + m*8 + ((m>>3)&1)*4 + ((k>>2)&3)*2 + (k&1)
#define A_H  160
#define A_CH 288
#define A_SZ (12 * A_CH)       // 3456 dwords
// B (64 cols x 192 K): idx = (k>>4)*B_CH + ((k>>1)&1)*B_H + n*8 + ((n>>3)&1)*4 + ((k>>2)&3)*2 + (k&1)
#define B_H  544
#define B_CH 1056
#define B_SZ (12 * B_CH)       // 12672 dwords

typedef __attribute__((ext_vector_type(2))) float v2f;
typedef __attribute__((ext_vector_type(4))) float v4f;
typedef __attribute__((ext_vector_type(8))) float v8f;

__device__ __forceinline__ float leaky(float x) { return x >= 0.f ? x : SLOPE * x; }

__device__ __forceinline__ unsigned encOrd(float f) {
    unsigned u = __float_as_uint(f);
    return (u & 0x80000000u) ? ~u : (u | 0x80000000u);
}
__device__ __forceinline__ float decOrd(unsigned u) {
    return (u & 0x80000000u) ? __uint_as_float(u ^ 0x80000000u) : __uint_as_float(~u);
}

__device__ __forceinline__ int aIdx(int k, int m) {
    return (k >> 4) * A_CH + ((k >> 1) & 1) * A_H + m * 8 + ((m >> 3) & 1) * 4 +
           ((k >> 2) & 3) * 2 + (k & 1);
}
__device__ __forceinline__ int bIdx(int k, int n) {
    return (k >> 4) * B_CH + ((k >> 1) & 1) * B_H + n * 8 + ((n >> 3) & 1) * 4 +
           ((k >> 2) & 3) * 2 + (k & 1);
}

#define WMMA_F32(cacc, ax, ay, bx, by)                                            \
    do {                                                                          \
        v2f aa, bb;                                                               \
        aa.x = (ax); aa.y = (ay); bb.x = (bx); bb.y = (by);                       \
        (cacc) = __builtin_amdgcn_wmma_f32_16x16x4_f32(false, aa, false, bb,      \
                                                       (short)0, (cacc), false, false); \
    } while (0)

// ---------------- init workspace ----------------
__global__ void init_ws_kernel(float* denom, unsigned* segMax,
                               float* sentAgg, float* recvAgg, int N) {
    int i = blockIdx.x * blockDim.x + threadIdx.x;
    int total = N * DD;
    if (i < total) { sentAgg[i] = 0.f; recvAgg[i] = 0.f; }
    if (i < N) { denom[i] = 0.f; segMax[i] = 0x007FFFFFu; /* enc(-inf) */ }
}

// ---------------- edge MLP + attention logits (WMMA) ----------------
__global__ void __launch_bounds__(128)
edge_mlp_attn_kernel(
    const float* __restrict__ nodes, const float* __restrict__ edges,
    const int* __restrict__ senders, const int* __restrict__ receivers,
    const float* __restrict__ Wedge, const float* __restrict__ bEdge,
    const float* __restrict__ Wattn, const float* __restrict__ bAttn,
    float* __restrict__ eNewG, float* __restrict__ logitsG,
    unsigned* __restrict__ segMax, int E)
{
    __shared__ __align__(16) float wC[B_SZ];
    __shared__ __align__(16) float xeC[A_SZ];
    __shared__ float wAttnLds[K3D];
    __shared__ float attnAcc[16];
    __shared__ int sIdx[16], rIdx[16];

    const int tid  = threadIdx.x;
    const int lane = tid & 31;
    const int wv   = tid >> 5;            // wave id -> N slice
    const int ln   = lane & 15;
    const int h    = lane >> 4;           // K-pair half
    const int n0   = wv * 16;
    const int nn   = n0 + ln;

    for (int i = tid; i < K3D * DD; i += 128) {
        int n = i / K3D, k = i - n * K3D;
        wC[bIdx(k, n)] = Wedge[i];
    }
    if (tid < K3D) wAttnLds[tid] = Wattn[tid];
    __syncthreads();

    const float bias = bEdge[nn];
    const float ba   = bAttn[0];
    const float wa_e = wAttnLds[2 * DD + nn];   // attn weight for this e_new column
    const int aBase  = h * A_H + ln * 8 + ((ln >> 3) & 1) * 4;
    const int bBase  = h * B_H + nn * 8 + ((nn >> 3) & 1) * 4;
    const int numTiles = (E + 15) >> 4;

    const int gm = tid >> 3;      // edge row this thread gathers
    const int kp = tid & 7;       // K sub-lane

    for (int tile = blockIdx.x; tile < numTiles; tile += gridDim.x) {
        const int e0 = tile << 4;
        if (tid < 16) {
            int e = e0 + tid;
            sIdx[tid] = (e < E) ? senders[e]   : 0;
            rIdx[tid] = (e < E) ? receivers[e] : 0;
            attnAcc[tid] = 0.f;
        }
        __syncthreads();

        // gather concat(sent, recv, edge) -> xeC; fuse attn partial for k<128
        {
            const int e = e0 + gm;
            const bool ok = (e < E);
            const int s = sIdx[gm], r = rIdx[gm];
            float pAcc = 0.f;
#pragma unroll
            for (int j = 0; j < 24; ++j) {
                int k = kp + j * 8;
                float v = 0.f;
                if (ok) {
                    if (k < DD)          v = nodes[(size_t)s * DD + k];
                    else if (k < 2 * DD) v = nodes[(size_t)r * DD + (k - DD)];
                    else                 v = edges[(size_t)e * DD + (k - 2 * DD)];
                }
                xeC[aIdx(k, gm)] = v;
                if (k < 2 * DD) pAcc += v * wAttnLds[k];
            }
            atomicAdd(&attnAcc[gm], pAcc);
        }
        __syncthreads();

        v8f c;
#pragma unroll
        for (int i = 0; i < 8; ++i) c[i] = bias;

#pragma unroll
        for (int ch = 0; ch < 12; ++ch) {
            v4f a0 = *(const v4f*)(xeC + aBase + ch * A_CH);
            v4f a1 = *(const v4f*)(xeC + aBase + ch * A_CH + 4);
            v4f b0 = *(const v4f*)(wC + bBase + ch * B_CH);
            v4f b1 = *(const v4f*)(wC + bBase + ch * B_CH + 4);
            WMMA_F32(c, a0.x, a0.y, b0.x, b0.y);
            WMMA_F32(c, a0.z, a0.w, b0.z, b0.w);
            WMMA_F32(c, a1.x, a1.y, b1.x, b1.y);
            WMMA_F32(c, a1.z, a1.w, b1.z, b1.w);
        }

        // leaky-relu, store e_new, accumulate e_new part of attention dot
        const int rb = h * 8;
#pragma unroll
        for (int r = 0; r < 8; ++r) {
            int m = rb + r;
            float v = leaky(c[r]);
            int e = e0 + m;
            if (e < E) eNewG[(size_t)e * DD + nn] = v;
            atomicAdd(&attnAcc[m], v * wa_e);
        }
        __syncthreads();

        if (tid < 16) {
            int e = e0 + tid;
            if (e < E) {
                float lg = leaky(attnAcc[tid] + ba);
                logitsG[e] = lg;
                atomicMax(&segMax[rIdx[tid]], encOrd(lg));
            }
        }
        __syncthreads();
    }
}

// ---------------- softmax numerator + denominator ----------------
__global__ void softmax_denom_kernel(const float* __restrict__ logitsG,
                                     const int* __restrict__ receivers,
                                     const unsigned* __restrict__ segMax,
                                     float* __restrict__ exBuf,
                                     float* __restrict__ denom, int E) {
    int e = blockIdx.x * blockDim.x + threadIdx.x;
    if (e >= E) return;
    int r = receivers[e];
    float ex = __expf(logitsG[e] - decOrd(segMax[r]));
    exBuf[e] = ex;
    atomicAdd(&denom[r], ex);
}

// ---------------- weight edges + scatter-add aggregations ----------------
__global__ void weight_scatter_kernel(const float* __restrict__ eNewG,
                                      const float* __restrict__ exBuf,
                                      const float* __restrict__ denom,
                                      const int* __restrict__ senders,
                                      const int* __restrict__ receivers,
                                      float* __restrict__ ewOut,
                                      float* __restrict__ sentAgg,
                                      float* __restrict__ recvAgg, int E) {
    int t = blockIdx.x * blockDim.x + threadIdx.x;
    int e = t >> 6;
    int c = t & 63;
    if (e >= E) return;
    int s = senders[e], r = receivers[e];
    float w = exBuf[e] / denom[r];
    float v = eNewG[(size_t)e * DD + c] * w;
    ewOut[(size_t)e * DD + c] = v;
    atomicAdd(&sentAgg[(size_t)s * DD + c], v);
    atomicAdd(&recvAgg[(size_t)r * DD + c], v);
}

// ---------------- node MLP (WMMA) ----------------
__global__ void __launch_bounds__(128)
node_mlp_kernel(const float* __restrict__ sentAgg,
                const float* __restrict__ recvAgg,
                const float* __restrict__ nodes,
                const float* __restrict__ Wnode,
                const float* __restrict__ bNode,
                float* __restrict__ nNew, int N)
{
    __shared__ __align__(16) float wC[B_SZ];
    __shared__ __align__(16) float xnC[A_SZ];

    const int tid  = threadIdx.x;
    const int lane = tid & 31;
    const int wv   = tid >> 5;
    const int ln   = lane & 15;
    const int h    = lane >> 4;
    const int n0   = wv * 16;
    const int nn   = n0 + ln;

    for (int i = tid; i < K3D * DD; i += 128) {
        int n = i / K3D, k = i - n * K3D;
        wC[bIdx(k, n)] = Wnode[i];
    }
    __syncthreads();

    const float bias = bNode[nn];
    const int aBase  = h * A_H + ln * 8 + ((ln >> 3) & 1) * 4;
    const int bBase  = h * B_H + nn * 8 + ((nn >> 3) & 1) * 4;
    const int numTiles = (N + 15) >> 4;

    const int gm = tid >> 3;
    const int kp = tid & 7;

    for (int tile = blockIdx.x; tile < numTiles; tile += gridDim.x) {
        const int m0 = tile << 4;
        {
            const int node = m0 + gm;
            const bool ok = (node < N);
#pragma unroll
            for (int j = 0; j < 24; ++j) {
                int k = kp + j * 8;
                float v = 0.f;
                if (ok) {
                    if (k < DD)          v = sentAgg[(size_t)node * DD + k];
                    else if (k < 2 * DD) v = recvAgg[(size_t)node * DD + (k - DD)];
                    else                 v = nodes[(size_t)node * DD + (k - 2 * DD)];
                }
                xnC[aIdx(k, gm)] = v;
            }
        }
        __syncthreads();

        v8f c;
#pragma unroll
        for (int i = 0; i < 8; ++i) c[i] = bias;

#pragma unroll
        for (int ch = 0; ch < 12; ++ch) {
            v4f a0 = *(const v4f*)(xnC + aBase + ch * A_CH);
            v4f a1 = *(const v4f*)(xnC + aBase + ch * A_CH + 4);
            v4f b0 = *(const v4f*)(wC + bBase + ch * B_CH);
            v4f b1 = *(const v4f*)(wC + bBase + ch * B_CH + 4);
            WMMA_F32(c, a0.x, a0.y, b0.x, b0.y);
            WMMA_F32(c, a0.z, a0.w, b0.z, b0.w);
            WMMA_F32(c, a1.x, a1.y, b1.x, b1.y);
            WMMA_F32(c, a1.z, a1.w, b1.z, b1.w);
        }

        const int rb = h * 8;
#pragma unroll
        for (int r = 0; r < 8; ++r) {
            int m = m0 + rb + r;
            if (m < N) nNew[(size_t)m * DD + nn] = leaky(c[r]);
        }
        __syncthreads();
    }
}

extern "C" void kernel_launch(void* const* d_in, const int* in_sizes, int n_in,
                              void* d_out, int out_size, void* d_ws, size_t ws_size,
                              hipStream_t stream) {
    const float* nodes     = (const float*)d_in[0];
    const float* edges     = (const float*)d_in[1];
    const int*   senders   = (const int*)d_in[2];
    const int*   receivers = (const int*)d_in[3];
    const float* Wedge     = (const float*)d_in[4];
    const float* bEdge     = (const float*)d_in[5];
    const float* Wnode     = (const float*)d_in[6];
    const float* bNode     = (const float*)d_in[7];
    const float* Wattn     = (const float*)d_in[8];
    const float* bAttn     = (const float*)d_in[9];

    const int N = in_sizes[0] / DD;
    const int E = in_sizes[2];

    // workspace layout
    float*    eNew    = (float*)d_ws;                 // E*64
    float*    logits  = eNew + (size_t)E * DD;        // E
    float*    exBuf   = logits + E;                   // E
    unsigned* segMax  = (unsigned*)(exBuf + E);       // N
    float*    denom   = (float*)(segMax + N);         // N
    float*    sentAgg = denom + N;                    // N*64
    float*    recvAgg = sentAgg + (size_t)N * DD;     // N*64

    // output layout: (n_new [N,64], e_w [E,64]) flattened in order
    float* nNewOut = (float*)d_out;
    float* ewOut   = nNewOut + (size_t)N * DD;

    init_ws_kernel<<<(N * DD + 255) / 256, 256, 0, stream>>>(denom, segMax, sentAgg, recvAgg, N);

    int numTilesE = (E + 15) / 16;
    int gridE = numTilesE < 2048 ? numTilesE : 2048;
    edge_mlp_attn_kernel<<<gridE, 128, 0, stream>>>(
        nodes, edges, senders, receivers, Wedge, bEdge, Wattn, bAttn,
        eNew, logits, segMax, E);

    softmax_denom_kernel<<<(E + 255) / 256, 256, 0, stream>>>(
        logits, receivers, segMax, exBuf, denom, E);

    weight_scatter_kernel<<<(E * 64 + 255) / 256, 256, 0, stream>>>(
        eNew, exBuf, denom, senders, receivers, ewOut, sentAgg, recvAgg, E);

    int numTilesN = (N + 15) / 16;
    int gridN = numTilesN < 2048 ? numTilesN : 2048;
    node_mlp_kernel<<<gridN, 128, 0, stream>>>(
        sentAgg, recvAgg, nodes, Wnode, bNode, nNewOut, N);
}